// Net_44049184588036
// MI455X (gfx1250) — compile-verified
//
#include <hip/hip_runtime.h>
#include <hip/hip_bf16.h>
#include <math.h>

typedef __attribute__((ext_vector_type(2))) float v2f;
typedef __attribute__((ext_vector_type(8))) float v8f;

// ---------------- zero workspace (deg + agg) ----------------
__global__ void __launch_bounds__(256) gcn_zero(float* __restrict__ p, int n) {
    int i = blockIdx.x * blockDim.x + threadIdx.x;
    if (i < n) p[i] = 0.0f;
}

// ---------------- in-degree via float atomics ----------------
__global__ void __launch_bounds__(256) gcn_deg(const int* __restrict__ col,
                                               float* __restrict__ deg, int E) {
    int i = blockIdx.x * blockDim.x + threadIdx.x;
    if (i < E) atomicAdd(&deg[col[i]], 1.0f);
}

// ---------------- dinv = rsqrt(deg+1), xs = x*dinv (in-place over deg) ----
__global__ void __launch_bounds__(256) gcn_dinv(float* __restrict__ deg_dinv,
                                                const float* __restrict__ x,
                                                float* __restrict__ xs, int n) {
    int i = blockIdx.x * blockDim.x + threadIdx.x;
    if (i < n) {
        float dv = rsqrtf(deg_dinv[i] + 1.0f);   // +1 = self loop; always > 0
        deg_dinv[i] = dv;
        xs[i] = x[i] * dv;
    }
}

// ---------------- layer-1 scalar scatter: agg[c] += x[r]*dinv[r] --------
__global__ void __launch_bounds__(256) gcn_scatter1(const int* __restrict__ row,
                                                    const int* __restrict__ col,
                                                    const float* __restrict__ xs,
                                                    float* __restrict__ agg, int E) {
    int i = blockIdx.x * blockDim.x + threadIdx.x;
    if (i < E) atomicAdd(&agg[col[i]], xs[row[i]]);
}

// ---------------- per-node dense: z = relu(s*W1+b1) @ W2 via WMMA -------
// One wave handles a 16-node tile. A = relu(H) (16x16 fp32), B = W2 padded
// to 16x16 (cols 2..15 zero). Chain of 4 x V_WMMA_F32_16X16X4_F32 over K.
// Writes zs[n][j] = z*dinv (gather source for pass 2) and initializes the
// output accumulator with the self-loop term (same value; dinv[c] applied
// in the finish pass).
__global__ void __launch_bounds__(256) gcn_node_wmma(
    const float* __restrict__ agg, const float* __restrict__ x,
    const float* __restrict__ dinv, const float* __restrict__ W1,
    const float* __restrict__ b1, const float* __restrict__ W2,
    float* __restrict__ zs, float* __restrict__ acc, int n) {

    int lane  = threadIdx.x & 31;
    int wave  = threadIdx.x >> 5;
    int tile  = blockIdx.x * 8 + wave;
    int base  = tile * 16;
    int m     = lane & 15;            // A row / B column within tile
    int khalf = (lane >> 4) << 1;     // 0 for lanes 0-15, 2 for lanes 16-31

    int node = base + m;
    node = node < n ? node : (n - 1); // clamp; out-of-range rows never stored
    float dv = dinv[node];
    float t  = dv * (agg[node] + x[node] * dv);   // layer-1 scalar pre-act

    v8f c = {0.f, 0.f, 0.f, 0.f, 0.f, 0.f, 0.f, 0.f};
#pragma unroll
    for (int kk = 0; kk < 16; kk += 4) {
        int ka = kk + khalf;
        v2f a, b;
        float h0 = t * W1[ka]     + b1[ka];
        float h1 = t * W1[ka + 1] + b1[ka + 1];
        a.x = h0 > 0.f ? h0 : 0.f;                 // relu
        a.y = h1 > 0.f ? h1 : 0.f;
        // B: lane = output column m; rows ka, ka+1; pad cols >= 2 with zero
        b.x = (m < 2) ? W2[ka * 2 + m]       : 0.f;
        b.y = (m < 2) ? W2[(ka + 1) * 2 + m] : 0.f;
        c = __builtin_amdgcn_wmma_f32_16x16x4_f32(
                false, a, false, b, (short)0, c, false, false);
    }

    // D layout: VGPR v -> row (lane<16 ? v : v+8), col = lane&15.
    if (m < 2) {
        int mb = (lane >> 4) * 8;
#pragma unroll
        for (int v = 0; v < 8; ++v) {
            int nd = base + mb + v;
            if (nd < n) {
                float zsv = c[v] * dinv[nd];       // z * dinv[node]
                zs[2 * nd + m]  = zsv;             // gather source
                acc[2 * nd + m] = zsv;             // self-loop init
            }
        }
    }
}

// ---------------- layer-2 float2 scatter: acc[c] += zs[r] ---------------
__global__ void __launch_bounds__(256) gcn_scatter2(const int* __restrict__ row,
                                                    const int* __restrict__ col,
                                                    const float* __restrict__ zs,
                                                    float* __restrict__ acc, int E) {
    int i = blockIdx.x * blockDim.x + threadIdx.x;
    if (i < E) {
        int r = row[i];
        int c = col[i];
        float z0 = zs[2 * r];
        float z1 = zs[2 * r + 1];
        atomicAdd(&acc[2 * c],     z0);
        atomicAdd(&acc[2 * c + 1], z1);
    }
}

// ---------------- finish: *dinv[c] + b2, softmax over 2 -----------------
__global__ void __launch_bounds__(256) gcn_finish(float* __restrict__ acc,
                                                  const float* __restrict__ dinv,
                                                  const float* __restrict__ b2, int n) {
    int i = blockIdx.x * blockDim.x + threadIdx.x;
    if (i < n) {
        float dv = dinv[i];
        float a0 = acc[2 * i]     * dv + b2[0];
        float a1 = acc[2 * i + 1] * dv + b2[1];
        float mx = fmaxf(a0, a1);
        float e0 = expf(a0 - mx);
        float e1 = expf(a1 - mx);
        float s  = e0 + e1;
        acc[2 * i]     = e0 / s;
        acc[2 * i + 1] = e1 / s;
    }
}

extern "C" void kernel_launch(void* const* d_in, const int* in_sizes, int n_in,
                              void* d_out, int out_size, void* d_ws, size_t ws_size,
                              hipStream_t stream) {
    const float* x  = (const float*)d_in[0];
    const int*   ei = (const int*)d_in[1];      // [2, E] int32
    const float* W1 = (const float*)d_in[2];    // [1,16]
    const float* b1 = (const float*)d_in[3];    // [16]
    const float* W2 = (const float*)d_in[4];    // [16,2]
    const float* b2 = (const float*)d_in[5];    // [2]

    const int n = in_sizes[0];                  // 250000
    const int E = in_sizes[1] / 2;              // 5000000
    const int* row = ei;
    const int* col = ei + E;

    float* ws   = (float*)d_ws;
    float* deg  = ws;                            // [n] -> becomes dinv
    float* agg  = ws + (size_t)n;                // [n]
    float* zsb  = ws + 2 * (size_t)n;            // [2n] (float2 per node)
    float* xs   = ws + 4 * (size_t)n;            // [n]
    float* acc  = (float*)d_out;                 // [2n] accumulator -> output

    const int B = 256;
    gcn_zero<<<(2 * n + B - 1) / B, B, 0, stream>>>(ws, 2 * n);          // deg+agg
    gcn_deg<<<(E + B - 1) / B, B, 0, stream>>>(col, deg, E);
    gcn_dinv<<<(n + B - 1) / B, B, 0, stream>>>(deg, x, xs, n);
    gcn_scatter1<<<(E + B - 1) / B, B, 0, stream>>>(row, col, xs, agg, E);

    int tiles  = (n + 15) / 16;                  // 16 nodes per wave
    int blocks = (tiles + 7) / 8;                // 8 waves per 256-thread block
    gcn_node_wmma<<<blocks, B, 0, stream>>>(agg, x, deg, W1, b1, W2, zsb, acc, n);

    gcn_scatter2<<<(E + B - 1) / B, B, 0, stream>>>(row, col, zsb, acc, E);
    gcn_finish<<<(n + B - 1) / B, B, 0, stream>>>(acc, deg, b2, n);
}